// standard_code_71347996721663
// MI455X (gfx1250) — compile-verified
//
#include <hip/hip_runtime.h>

// Elementwise masked-threshold, bandwidth-bound (~384 MB moved, ~16.5 us floor
// at 23.3 TB/s). No matrix contraction -> no WMMA. Strategy: b128 NT streaming;
// params staged via CDNA5 async-to-LDS DMA (ASYNCcnt), then hoisted into VGPRs
// (the thread's param block index is loop-invariant under the grid stride).

typedef __attribute__((ext_vector_type(4))) float f32x4;
typedef __attribute__((__vector_size__(4 * sizeof(int)))) int v4i;
typedef __attribute__((address_space(1))) v4i* gv4i_p;   // global b128
typedef __attribute__((address_space(3))) v4i* lv4i_p;   // LDS b128

#define THREADS 256
#define D_FEAT 512
#define VEC 4
#define FB (D_FEAT / VEC)   // 128 float4 param blocks per row

static_assert(FB == 128, "index math below assumes 128 float4 blocks per row");
static_assert(THREADS % FB == 0, "grid stride must be a multiple of FB");

__device__ __forceinline__ void apply_thresh(float xv, float mu, float sg,
                                             float& m, float& s) {
  const float one = 1.0f, zero = 0.0f;
  const float a = sg + mu;   // upper threshold
  const float b = sg - mu;   // lower threshold
  // mean channel: sequential (x>mean -> 1) then (<mean -> 0)
  float t = (xv > mu) ? one : xv;
  m = (t < mu) ? zero : t;
  // sigma channel: four sequential masked assigns
  float u = (xv > a) ? one : xv;
  u = (u < b) ? one : u;
  u = (u < a) ? zero : u;
  s = (u > b) ? zero : u;
}

__global__ __launch_bounds__(THREADS) void thresh_interleave_kernel(
    const float* __restrict__ x, const float* __restrict__ mean,
    const float* __restrict__ sigma, float* __restrict__ out,
    long long total_vec) {
  __shared__ f32x4 sm_mean[FB];
  __shared__ f32x4 sm_sig[FB];
  const int tid = threadIdx.x;

#if __has_builtin(__builtin_amdgcn_global_load_async_to_lds_b128)
  // CDNA5 async DMA global->LDS (ASYNCcnt); one b128 per lane for each table.
  if (tid < FB) {
    __builtin_amdgcn_global_load_async_to_lds_b128(
        (gv4i_p)(mean + 4 * tid), (lv4i_p)(&sm_mean[tid]),
        /*offset=*/0, /*cpol=*/0);
    __builtin_amdgcn_global_load_async_to_lds_b128(
        (gv4i_p)(sigma + 4 * tid), (lv4i_p)(&sm_sig[tid]),
        /*offset=*/0, /*cpol=*/0);
  }
  asm volatile("s_wait_asynccnt 0" ::: "memory");
#else
  if (tid < FB) {
    sm_mean[tid] = reinterpret_cast<const f32x4*>(mean)[tid];
    sm_sig[tid] = reinterpret_cast<const f32x4*>(sigma)[tid];
  }
#endif
  __syncthreads();

  const f32x4* __restrict__ x4 = reinterpret_cast<const f32x4*>(x);
  f32x4* __restrict__ out4 = reinterpret_cast<f32x4*>(out);
  const long long stride = (long long)gridDim.x * THREADS;  // multiple of FB
  const long long g0 = (long long)blockIdx.x * THREADS + tid;
  if (g0 >= total_vec) return;

  // stride % FB == 0  =>  (g & (FB-1)) is invariant: read params once.
  const int fb = (int)(g0 & (FB - 1));
  const f32x4 mv = sm_mean[fb];   // one ds_load_b128 each, held in VGPRs
  const f32x4 sv = sm_sig[fb];

  for (long long g = g0; g < total_vec; g += stride) {
    // Streaming read of x: non-temporal (never re-read).
    f32x4 xv = __builtin_nontemporal_load(&x4[g]);

    float m0, s0, m1, s1, m2, s2, m3, s3;
    apply_thresh(xv.x, mv.x, sv.x, m0, s0);
    apply_thresh(xv.y, mv.y, sv.y, m1, s1);
    apply_thresh(xv.z, mv.z, sv.z, m2, s2);
    apply_thresh(xv.w, mv.w, sv.w, m3, s3);

    f32x4 o0 = {m0, s0, m1, s1};   // interleaved (mean_i, sigma_i) pairs
    f32x4 o1 = {m2, s2, m3, s3};

    // out float4 index = 2*g  (row*256 + 2*fb == 2*(row*128 + fb))
    const long long ob = g << 1;
    // Streaming writes: non-temporal so the 256 MB output doesn't evict L2.
    __builtin_nontemporal_store(o0, &out4[ob]);
    __builtin_nontemporal_store(o1, &out4[ob + 1]);
  }
}

extern "C" void kernel_launch(void* const* d_in, const int* in_sizes, int n_in,
                              void* d_out, int out_size, void* d_ws,
                              size_t ws_size, hipStream_t stream) {
  (void)n_in; (void)out_size; (void)d_ws; (void)ws_size;
  const float* x = (const float*)d_in[0];
  const float* mean = (const float*)d_in[1];
  const float* sigma = (const float*)d_in[2];
  float* out = (float*)d_out;

  const long long n_elems = (long long)in_sizes[0];  // N * D
  const long long total_vec = n_elems / VEC;         // N * 128

  long long needed = (total_vec + THREADS - 1) / THREADS;
  int blocks = 4096;  // grid-stride: 8 iters/thread at N=65536, D=512
  if (needed < blocks) blocks = (int)needed;
  if (blocks < 1) blocks = 1;

  hipLaunchKernelGGL(thresh_interleave_kernel, dim3(blocks), dim3(THREADS), 0,
                     stream, x, mean, sigma, out, total_vec);
}